// AlignedContrastiveLoss_72636486910296
// MI455X (gfx1250) — compile-verified
//
#include <hip/hip_runtime.h>
#include <hip/hip_bf16.h>

typedef float v2f __attribute__((ext_vector_type(2)));
typedef float v8f __attribute__((ext_vector_type(8)));

#define BB    8
#define CC    256
#define HH    128
#define WW    128
#define HWSZ  (HH * WW)
#define NTOPK 10
#define NROW  16
#define NEG_INF (-__builtin_inff())

// ---------------------------------------------------------------------------
// K1: intensity[map][b][h][w] = || f[b][:,h,w] ||_2  (map 0 = det, 1 = loc)
// 4 pixels per thread -> float4 (B128) coalesced loads; this streams 268 MB
// and is the only bandwidth-relevant kernel (~11.5 us at 23.3 TB/s).
// ---------------------------------------------------------------------------
__global__ void k_intensity(const float* __restrict__ det,
                            const float* __restrict__ loc,
                            float* __restrict__ inten) {
  const float* src = (blockIdx.y == 0) ? det : loc;
  const int g   = blockIdx.x * blockDim.x + threadIdx.x;  // group of 4 pixels
  const int pix = g * 4;
  const int b   = pix / HWSZ;
  const int hw  = pix % HWSZ;
  const float* p = src + (size_t)b * CC * HWSZ + hw;
  float ax = 0.f, ay = 0.f, az = 0.f, aw = 0.f;
#pragma unroll 4
  for (int c = 0; c < CC; ++c) {
    const float4 v = *reinterpret_cast<const float4*>(p + (size_t)c * HWSZ);
    ax = fmaf(v.x, v.x, ax);
    ay = fmaf(v.y, v.y, ay);
    az = fmaf(v.z, v.z, az);
    aw = fmaf(v.w, v.w, aw);
  }
  float4 o;
  o.x = sqrtf(ax); o.y = sqrtf(ay); o.z = sqrtf(az); o.w = sqrtf(aw);
  *reinterpret_cast<float4*>(inten + (size_t)blockIdx.y * BB * HWSZ + pix) = o;
}

// ---------------------------------------------------------------------------
// K2: masked[i] = (I[i] == max3x3(I)[i] && I[i] > 0.5) ? I[i] : -inf
// SAME padding with -inf identity == clamped window (self always included).
// ---------------------------------------------------------------------------
__global__ void k_peaks(const float* __restrict__ inten,
                        float* __restrict__ masked) {
  const int idx = blockIdx.x * blockDim.x + threadIdx.x;  // 0 .. 2*B*HW-1
  const int mb  = idx / HWSZ;
  const int hw  = idx % HWSZ;
  const int y = hw / WW, x = hw % WW;
  const float* I = inten + (size_t)mb * HWSZ;
  const float v = I[hw];
  const int y0 = (y > 0) ? y - 1 : y, y1 = (y < HH - 1) ? y + 1 : y;
  const int x0 = (x > 0) ? x - 1 : x, x1 = (x < WW - 1) ? x + 1 : x;
  float m = v;
  for (int yy = y0; yy <= y1; ++yy)
    for (int xx = x0; xx <= x1; ++xx)
      m = fmaxf(m, I[yy * WW + xx]);
  masked[idx] = ((v == m) && (v > 0.5f)) ? v : NEG_INF;
}

// ---------------------------------------------------------------------------
// K3: destructive top-10 per map (value desc, tie -> lowest index, matching
// lax.top_k). Pads slots 10..15 with -inf so later stages treat rows >= 10
// as invalid uniformly.
// ---------------------------------------------------------------------------
__global__ void k_topk(float* __restrict__ masked,
                       float* __restrict__ vals,
                       int* __restrict__ inds) {
  __shared__ float sv[256];
  __shared__ int   si[256];
  const int mb  = blockIdx.x;
  const int tid = threadIdx.x;
  float* M = masked + (size_t)mb * HWSZ;

  for (int k = 0; k < NTOPK; ++k) {
    float best = NEG_INF;
    int   bi   = HWSZ;
    for (int i = tid; i < HWSZ; i += 256) {
      const float v = M[i];
      if (v > best) { best = v; bi = i; }  // ascending i -> lowest index kept
    }
    sv[tid] = best; si[tid] = bi;
    __syncthreads();
    for (int s = 128; s > 0; s >>= 1) {
      if (tid < s) {
        const float v2 = sv[tid + s];
        const int   i2 = si[tid + s];
        if (v2 > sv[tid] || (v2 == sv[tid] && i2 < si[tid])) {
          sv[tid] = v2; si[tid] = i2;
        }
      }
      __syncthreads();
    }
    if (tid == 0) {
      vals[(size_t)mb * NROW + k] = sv[0];
      inds[(size_t)mb * NROW + k] = (si[0] < HWSZ) ? si[0] : 0;
      if (si[0] < HWSZ) M[si[0]] = NEG_INF;
    }
    __syncthreads();
  }
  if (tid >= NTOPK && tid < NROW) {
    vals[(size_t)mb * NROW + tid] = NEG_INF;
    inds[(size_t)mb * NROW + tid] = 0;
  }
}

// ---------------------------------------------------------------------------
// K4: gather feats[r][c] = f[b][c][peak_r] (0 for invalid rows) and
// norms[r] = max(||feats[r]||, 1e-8). One block per (map, b, row).
// ---------------------------------------------------------------------------
__global__ void k_gather(const float* __restrict__ det,
                         const float* __restrict__ loc,
                         const float* __restrict__ vals,
                         const int* __restrict__ inds,
                         float* __restrict__ feats,
                         float* __restrict__ norms) {
  __shared__ float red[256];
  const int r   = blockIdx.x;       // 0 .. 2*B*16-1
  const int mb  = r / NROW;
  const int row = r % NROW;
  const int map = mb / BB;
  const int b   = mb % BB;
  const float* src = (map == 0) ? det : loc;
  const int c = threadIdx.x;

  const float v    = vals[(size_t)mb * NROW + row];
  const bool valid = (row < NTOPK) && (v > NEG_INF);
  const int  hw    = inds[(size_t)mb * NROW + row];
  const float f = valid ? src[(size_t)b * CC * HWSZ + (size_t)c * HWSZ + hw] : 0.0f;
  feats[(size_t)r * CC + c] = f;

  red[c] = f * f;
  __syncthreads();
  for (int s = 128; s > 0; s >>= 1) {
    if (c < s) red[c] += red[c + s];
    __syncthreads();
  }
  if (c == 0) norms[r] = fmaxf(sqrtf(red[0]), 1e-8f);
}

// ---------------------------------------------------------------------------
// K5: per-sample loss. One wave32 per sample; full-f32 Gram matrix
// det_f(16x256) x loc_f(16x256)^T via 64 chained V_WMMA_F32_16X16X4_F32.
// ISA 7.12.2 layouts:
//   A 16x4:  lane l -> M = l%16; VGPR j holds K = 2*(l>=16) + j
//   B 4x16:  lane l -> N = l%16; VGPR j holds K = 2*(l>=16) + j (mirrored)
//   D 16x16: lane l, VGPR v -> (M = v + 8*(l>=16), N = l%16)
// ---------------------------------------------------------------------------
__global__ void k_loss(const float* __restrict__ feats,
                       const float* __restrict__ norms,
                       const float* __restrict__ vals,
                       float* __restrict__ losses) {
  const int b    = blockIdx.x;
  const int lane = threadIdx.x;                 // 0..31, EXEC all ones
  const float* A  = feats + (size_t)(0 * BB + b) * NROW * CC;  // det rows
  const float* Bm = feats + (size_t)(1 * BB + b) * NROW * CC;  // loc rows

  const int mrow  = lane & 15;
  const int khalf = (lane >> 4) << 1;           // 0 or 2
  const float* arow = A  + mrow * CC + khalf;
  const float* brow = Bm + mrow * CC + khalf;

  v8f acc = {};
#pragma unroll 8
  for (int k0 = 0; k0 < CC; k0 += 4) {
    const v2f av = *reinterpret_cast<const v2f*>(arow + k0);
    const v2f bv = *reinterpret_cast<const v2f*>(brow + k0);
    acc = __builtin_amdgcn_wmma_f32_16x16x4_f32(
        /*neg_a=*/false, av, /*neg_b=*/false, bv,
        /*c_mod=*/(short)0, acc, /*reuse_a=*/false, /*reuse_b=*/false);
  }

  const float* valsD = vals + (size_t)(0 * BB + b) * NROW;
  const float* valsL = vals + (size_t)(1 * BB + b) * NROW;
  const float* nD    = norms + (size_t)(0 * BB + b) * NROW;
  const float* nL    = norms + (size_t)(1 * BB + b) * NROW;

  int cntD = 0, cntL = 0;
#pragma unroll
  for (int i = 0; i < NROW; ++i) {
    cntD += (valsD[i] > NEG_INF) ? 1 : 0;
    cntL += (valsL[i] > NEG_INF) ? 1 : 0;
  }

  const int   n     = lane & 15;
  const int   mbase = (lane < 16) ? 0 : 8;
  const float nb_   = nL[n];
  const bool  vL    = (valsL[n] > NEG_INF);

  float partial = 0.f;
#pragma unroll
  for (int v = 0; v < 8; ++v) {
    const int  m   = mbase + v;
    const bool ok  = vL && (valsD[m] > NEG_INF);
    const float sim = acc[v] / (nD[m] * nb_);
    const float r   = sim - 0.5f;
    partial += (ok && (r > 0.f)) ? r : 0.f;
  }
  for (int off = 16; off > 0; off >>= 1)
    partial += __shfl_down(partial, off, 32);

  if (lane == 0) {
    const long long npairs = (long long)cntD * (long long)cntL;
    losses[b] = (npairs > 0) ? (partial / (float)npairs) : 0.0f;
  }
}

// K6: final fixed-order batch mean -> d_out[0] (rewritten every call).
__global__ void k_final(const float* __restrict__ losses, float* __restrict__ out) {
  if (threadIdx.x == 0 && blockIdx.x == 0) {
    float s = 0.f;
    for (int i = 0; i < BB; ++i) s += losses[i];
    out[0] = s / (float)BB;
  }
}

extern "C" void kernel_launch(void* const* d_in, const int* in_sizes, int n_in,
                              void* d_out, int out_size, void* d_ws, size_t ws_size,
                              hipStream_t stream) {
  (void)in_sizes; (void)n_in; (void)out_size; (void)ws_size;
  const float* loc = (const float*)d_in[0];   // loc_features [8,256,128,128]
  const float* det = (const float*)d_in[1];   // det_features [8,256,128,128]
  float* out = (float*)d_out;

  // Workspace carve-up (~2.4 MB total)
  float* inten  = (float*)d_ws;                    // 2*B*HW
  float* masked = inten  + 2 * BB * HWSZ;          // 2*B*HW
  float* vals   = masked + 2 * BB * HWSZ;          // 2*B*16
  int*   inds   = (int*)(vals + 2 * BB * NROW);    // 2*B*16
  float* feats  = (float*)(inds + 2 * BB * NROW);  // 2*B*16*256
  float* norms  = feats  + 2 * BB * NROW * CC;     // 2*B*16
  float* losses = norms  + 2 * BB * NROW;          // B

  k_intensity<<<dim3((BB * HWSZ / 4) / 256, 2), 256, 0, stream>>>(det, loc, inten);
  k_peaks   <<<(2 * BB * HWSZ) / 256, 256, 0, stream>>>(inten, masked);
  k_topk    <<<2 * BB, 256, 0, stream>>>(masked, vals, inds);
  k_gather  <<<2 * BB * NROW, 256, 0, stream>>>(det, loc, vals, inds, feats, norms);
  k_loss    <<<BB, 32, 0, stream>>>(feats, norms, vals, losses);
  k_final   <<<1, 32, 0, stream>>>(losses, out);
}